// ParallelSynapseLayer_40965398069703
// MI455X (gfx1250) — compile-verified
//
#include <hip/hip_runtime.h>

// Problem dims (fixed by the reference)
#define N_DATA  128
#define N_SYN   8
#define IN_DIM  784
#define OUT_DIM 128
#define I_CHUNK 49          // 16 chunks * 49 = 784

typedef __attribute__((ext_vector_type(2))) float v2f;
typedef __attribute__((ext_vector_type(8))) float v8f;

__device__ __forceinline__ float fast_tanh(float v) {
#if __has_builtin(__builtin_amdgcn_tanhf)
    return __builtin_amdgcn_tanhf(v);
#else
    float r;
    // v_tanh_f32 is a TRANS op: 1 independent op / v_nop required before use.
    asm("v_tanh_f32 %0, %1\n\tv_nop" : "=v"(r) : "v"(v));
    return r;
#endif
}

// One wave (32 lanes) per workgroup. Each wave owns a 16n x 16o output tile
// for one synapse s and one 49-wide i-chunk; partials are atomically added.
//
// WMMA f32 16x16x4 slot mapping (ISA 7.12.2, consistent for A and B):
//   slot(half, vgpr) -> K = 2*half + vgpr
// We use K=0 for x*slope and K=1 for 1*(-thres); K=2,3 rows of B are zero,
// so D = x[n]*slope[o] - thres[o] with C = 0 (loop-invariant).
__global__ __launch_bounds__(32)
void psyn_wmma_kernel(const float* __restrict__ x,
                      const float* __restrict__ thres,
                      const float* __restrict__ slope,
                      const float* __restrict__ ampli,
                      float* __restrict__ out) {
    __shared__ float lds_x[I_CHUNK * 16];   // [i_local][n] : 3136 B

    const int lane  = threadIdx.x;   // 0..31
    const int oc    = lane & 15;     // o sub-index (also n sub-index for A)
    const int half  = lane >> 4;     // 0 = lanes 0-15, 1 = lanes 16-31
    const bool low  = (half == 0);

    const int tile = blockIdx.x;            // 0..63
    const int n0   = (tile & 7) * 16;
    const int o0   = (tile >> 3) * 16;
    const int s    = blockIdx.y;            // 0..7
    const int i0   = blockIdx.z * I_CHUNK;  // 0..15 chunks

    // Stage x tile (16 n x 49 i) into LDS, transposed to [il][n] so the
    // per-iteration read lds_x[il*16+oc] is a conflict-free 16-bank broadcast.
    for (int t = lane; t < I_CHUNK * 16; t += 32) {
        const int n  = t & 15;
        const int il = t >> 4;
        lds_x[il * 16 + n] = x[(n0 + n) * IN_DIM + (i0 + il)];
    }
    __syncthreads();  // single-wave WG: collapses; cnt waits still enforced

    // Per-lane streaming pointers; advance by one o-row (OUT_DIM floats).
    const size_t base0 = (size_t)(s * IN_DIM + i0) * OUT_DIM + (size_t)(o0 + oc);
    const float* __restrict__ sp = slope + base0;
    const float* __restrict__ tp = thres + base0;
    const float* __restrict__ ap = ampli + base0;

    const v8f czero = {0.f, 0.f, 0.f, 0.f, 0.f, 0.f, 0.f, 0.f};
    v8f acc = czero;

    for (int il = 0; il < I_CHUNK; ++il) {
        const float sl = *sp;  sp += OUT_DIM;   // coalesced 16 lanes + bcast
        const float th = *tp;  tp += OUT_DIM;
        const float am = *ap;  ap += OUT_DIM;
        const float w  = am * am;
        const float xv = lds_x[il * 16 + oc];

        // A: K=0,2 slots = x[n]; K=1,3 slots = 1.0 (bias multiplier).
        v2f a; a.x = xv; a.y = 1.0f;
        // B: K=0 row = slope[o]; K=1 row = -thres[o]; K=2,3 rows = 0.
        v2f b;
        b.x = low ?  sl : 0.0f;
        b.y = low ? -th : 0.0f;

        // z[n,o] = slope*x - thres for the whole 16x16 tile, one instruction.
        v8f z = __builtin_amdgcn_wmma_f32_16x16x4_f32(
            false, a, false, b, (short)0, czero, false, false);

        // Trans-pipe bottleneck: 8x hardware tanh, then dual-issued FMAs.
        #pragma unroll
        for (int r = 0; r < 8; ++r)
            acc[r] += fast_tanh(z[r]) * w;
    }

    // C/D layout: VGPR r -> row M = r (lanes 0-15) / r+8 (lanes 16-31).
    const int ocol = o0 + oc;
    #pragma unroll
    for (int r = 0; r < 8; ++r) {
        const int row = n0 + r + half * 8;
        __hip_atomic_fetch_add(&out[row * OUT_DIM + ocol], acc[r],
                               __ATOMIC_RELAXED, __HIP_MEMORY_SCOPE_AGENT);
    }
}

extern "C" void kernel_launch(void* const* d_in, const int* in_sizes, int n_in,
                              void* d_out, int out_size, void* d_ws, size_t ws_size,
                              hipStream_t stream) {
    (void)in_sizes; (void)n_in; (void)d_ws; (void)ws_size;
    const float* x     = (const float*)d_in[0];
    const float* thres = (const float*)d_in[1];
    const float* slope = (const float*)d_in[2];
    const float* ampli = (const float*)d_in[3];
    float* out = (float*)d_out;

    // Zero the accumulation target (graph-capturable async memset).
    hipMemsetAsync(d_out, 0, (size_t)out_size * sizeof(float), stream);

    // 64 (n,o) tiles  x  8 synapses  x  16 i-chunks = 8192 single-wave WGs.
    dim3 grid(64, N_SYN, IN_DIM / I_CHUNK);
    psyn_wmma_kernel<<<grid, 32, 0, stream>>>(x, thres, slope, ampli, out);
}